// Net_45363444580661
// MI455X (gfx1250) — compile-verified
//
#include <hip/hip_runtime.h>

// Allen-Cahn stencil step, NCHW 16x1x1024x1024 fp32.
// Bandwidth-bound (AI ~1.3 flop/B): optimize the data path, not the math.
// Uses CDNA5 async global->LDS loads (ASYNCcnt) to stage halo tiles.

#define IMG_H 1024
#define IMG_W 1024
#define TX 128          // tile width  (outputs)
#define TY 16           // tile height (outputs)
#define HALO_W (TX + 2) // 130
#define HALO_H (TY + 2) // 18
#define TXP 132         // padded LDS row stride (floats)
#define NLOAD (HALO_W * HALO_H) // 2340 halo elements per tile

#define ALPHA 400.0f    // 1/eps^2
#define BETA  10000.0f  // 1/h^2

__global__ __launch_bounds__(256) void allen_cahn_kernel(
    const float* __restrict__ x0, float* __restrict__ out)
{
    __shared__ float smem[HALO_H * TXP];

    const int tid  = threadIdx.x;
    const int tx0  = blockIdx.x * TX;
    const int ty0  = blockIdx.y * TY;
    const size_t plane = (size_t)blockIdx.z * (size_t)(IMG_H * IMG_W);
    const float* src = x0 + plane;

    // ---- Stage (TY+2)x(TX+2) halo tile into LDS via async global->LDS ----
    // Per-lane clamped global address implements edge replication for free.
    for (int i = tid; i < NLOAD; i += 256) {
        const int ly = i / HALO_W;
        const int lx = i - ly * HALO_W;
        int gy = ty0 + ly - 1;
        gy = gy < 0 ? 0 : (gy > IMG_H - 1 ? IMG_H - 1 : gy);
        int gx = tx0 + lx - 1;
        gx = gx < 0 ? 0 : (gx > IMG_W - 1 ? IMG_W - 1 : gx);

        const float* gp = src + (size_t)gy * IMG_W + gx;
        unsigned lds_addr = (unsigned)(size_t)(&smem[ly * TXP + lx]);

        // GV mode: vdst gives the LDS address, vaddr the 64-bit global address; tracked with ASYNCcnt.
        asm volatile("global_load_async_to_lds_b32 %0, %1, off"
                     :
                     : "v"(lds_addr), "v"(gp)
                     : "memory");
    }
    // Each wave drains its own async ops, then the barrier makes all waves'
    // LDS writes visible to everyone.
    asm volatile("s_wait_asynccnt 0" ::: "memory");
    __syncthreads();

    // ---- Compute 5-point stencil + cubic reaction from LDS ----
    const int lane = tid & 31;
    const int wv   = tid >> 5;          // 0..7
    const int xb   = lane * 4;          // column group within tile: 0..124
    float* dst = out + plane;

#pragma unroll
    for (int rr = 0; rr < 2; ++rr) {
        const int row = wv + rr * 8;    // 0..15
        const float* rN = &smem[(row + 0) * TXP];
        const float* rC = &smem[(row + 1) * TXP];
        const float* rS = &smem[(row + 2) * TXP];

        float4 res;
        float* resp = &res.x;
#pragma unroll
        for (int j = 0; j < 4; ++j) {
            const int hx = xb + 1 + j;  // halo-coordinate center column
            const float c = rC[hx];
            const float n = rN[hx];
            const float s = rS[hx];
            const float w = rC[hx - 1];
            const float e = rC[hx + 1];
            const float diffusion = (n + s) + (w + e) - 4.0f * c;
            resp[j] = ALPHA * c - ALPHA * (c * c * c) + BETA * diffusion;
        }

        // tx0 is a multiple of 128 and xb of 4 -> 16B-aligned float4 store.
        *reinterpret_cast<float4*>(
            dst + (size_t)(ty0 + row) * IMG_W + tx0 + xb) = res;
    }
}

extern "C" void kernel_launch(void* const* d_in, const int* in_sizes, int n_in,
                              void* d_out, int out_size, void* d_ws, size_t ws_size,
                              hipStream_t stream)
{
    (void)n_in; (void)out_size; (void)d_ws; (void)ws_size;
    const float* x0 = (const float*)d_in[0];
    float* out      = (float*)d_out;

    const int batches = in_sizes[0] / (IMG_H * IMG_W);   // 16
    dim3 grid(IMG_W / TX, IMG_H / TY, batches);
    allen_cahn_kernel<<<grid, dim3(256), 0, stream>>>(x0, out);
}